// UpdateRule_82085414961361
// MI455X (gfx1250) — compile-verified
//
#include <hip/hip_runtime.h>
#include <math.h>

// ---------------------------------------------------------------------------
// GCN(max-aggr) x2 + residual tanh for MI455X (gfx1250, wave32).
//   GEMMs via V_WMMA_F32_16X16X4_F32 (fp32-exact matrix path), branch-free
//   inner loop (guards only in a 2-wide masked tail for K=130).
//   segment_max via order-preserving fp32<->u32 encoding + global_atomic_max_u32.
//   Both N*128 fp32 scratch buffers (51.2MB each) fit in the 192MB L2 together,
//   so the edge-scatter phase is L2-resident.
// Workspace: 2 * N * 128 * 4 bytes = 102.4 MB.
// ---------------------------------------------------------------------------

typedef __attribute__((ext_vector_type(2))) float v2f;
typedef __attribute__((ext_vector_type(8))) float v8f;

// Order-preserving map: fp32 compare == u32 compare on encoded values.
__device__ __forceinline__ unsigned f2ord(float f) {
    unsigned u = __float_as_uint(f);
    return (u & 0x80000000u) ? ~u : (u | 0x80000000u);
}
__device__ __forceinline__ float ord2f(unsigned u) {
    unsigned b = (u & 0x80000000u) ? (u ^ 0x80000000u) : ~u;
    return __uint_as_float(b);
}

// ---------------------------------------------------------------------------
// C[M x 128] = A[M x K] * B[K x 128], fp32, one wave per 16-row slab.
// Wave holds 8 accumulator tiles (16x16) covering all 128 columns; the A
// fragment is one aligned b64 load per k-step, reused across 8 WMMAs.
// Requires K even (true here: 130, 128). M-tail rows clamped on load,
// guarded on store.
// WMMA_F32_16X16X4_F32 layouts (wave32):
//   A 16x4 : lanes 0-15 rows 0-15 {K=0,1}, lanes 16-31 rows 0-15 {K=2,3}
//   B 4x16 : lanes 0-15 cols 0-15 {K=0,1}, lanes 16-31 cols 0-15 {K=2,3}
//   C 16x16: VGPR v -> row v (lanes 0-15) / row v+8 (lanes 16-31), col = lane&15
// ---------------------------------------------------------------------------
__global__ __launch_bounds__(256) void gemm_wmma_f32(
        const float* __restrict__ A, const float* __restrict__ B,
        float* __restrict__ C, int M, int K) {
    const int NCOL = 128;
    int wave = (blockIdx.x << 3) | (threadIdx.x >> 5);   // 8 waves / block
    int m0 = wave << 4;
    if (m0 >= M) return;                                  // wave-uniform exit

    int lane = threadIdx.x & 31;
    int r    = lane & 15;        // row (A) / col (B) within tile
    int half = lane >> 4;        // selects K pair {0,1} vs {2,3}

    v8f acc[8] = {};             // 8 x (16x16 f32) = full 128 columns

    int arow = m0 + r;
    if (arow >= M) arow = M - 1;                          // tail clamp (loads only)
    const float* __restrict__ Ar = A + (size_t)arow * K;

    const int Kmain = K & ~3;

    for (int k = 0; k < Kmain; k += 4) {
        int k0 = k + (half << 1);
        v2f a = *(const v2f*)(Ar + k0);                   // aligned b64 load
        const float* __restrict__ Bp = B + ((size_t)k0 * NCOL) + r;
        float b0[8], b1[8];
#pragma unroll
        for (int t = 0; t < 8; ++t) {                     // batch B loads first
            b0[t] = Bp[t << 4];
            b1[t] = Bp[NCOL + (t << 4)];
        }
#pragma unroll
        for (int t = 0; t < 8; ++t) {
            v2f b; b.x = b0[t]; b.y = b1[t];
            acc[t] = __builtin_amdgcn_wmma_f32_16x16x4_f32(
                false, a, false, b, (short)0, acc[t], false, false);
        }
    }

    if (Kmain < K) {
        // 2 leftover k-values (K % 4 == 2). All lanes load the valid rows
        // Kmain/Kmain+1; half==1 lanes (K slots 2,3) zero their A fragment via
        // multiplicative mask -> their a*b products vanish. No branches.
        float m = half ? 0.0f : 1.0f;
        v2f a = *(const v2f*)(Ar + Kmain);
        a.x *= m; a.y *= m;
        const float* __restrict__ Bp = B + ((size_t)Kmain * NCOL) + r;
        float b0[8], b1[8];
#pragma unroll
        for (int t = 0; t < 8; ++t) {
            b0[t] = Bp[t << 4];
            b1[t] = Bp[NCOL + (t << 4)];
        }
#pragma unroll
        for (int t = 0; t < 8; ++t) {
            v2f b; b.x = b0[t]; b.y = b1[t];
            acc[t] = __builtin_amdgcn_wmma_f32_16x16x4_f32(
                false, a, false, b, (short)0, acc[t], false, false);
        }
    }

#pragma unroll
    for (int t = 0; t < 8; ++t) {
#pragma unroll
        for (int v = 0; v < 8; ++v) {
            int row = m0 + (half << 3) + v;
            if (row < M) C[(size_t)row * NCOL + (t << 4) + r] = acc[t][v];
        }
    }
}

// Encode transformed features to ordered-u32 in place (src of gathers) and
// copy into the aggregation buffer (this IS the self-loop contribution).
__global__ __launch_bounds__(256) void encode_self(
        const float* __restrict__ h, unsigned* __restrict__ henc,
        unsigned* __restrict__ agg, int n) {
    int i = blockIdx.x * 256 + threadIdx.x;
    if (i < n) {
        unsigned e = f2ord(h[i]);
        henc[i] = e;   // henc aliases h (in-place re-encode)
        agg[i]  = e;
    }
}

// One wave per edge: lane loads uint4 (4 features) of henc[src], 4 atomic max
// into agg[dst]. 32 lanes * 4 = 128 features. Everything L2-resident.
__global__ __launch_bounds__(256) void scatter_max_kernel(
        const unsigned* __restrict__ henc, const int* __restrict__ src,
        const int* __restrict__ dst, unsigned* __restrict__ agg, int E) {
    int e = (blockIdx.x << 3) | (threadIdx.x >> 5);
    if (e >= E) return;                                   // wave-uniform exit
    int s = __builtin_amdgcn_readfirstlane(src[e]);       // uniform -> SGPR base
    int d = __builtin_amdgcn_readfirstlane(dst[e]);
    int lane = threadIdx.x & 31;

    const uint4 v = ((const uint4*)(henc + (size_t)s * 128))[lane];
    unsigned* out = agg + (size_t)d * 128 + lane * 4;
    atomicMax(out + 0, v.x);
    atomicMax(out + 1, v.y);
    atomicMax(out + 2, v.z);
    atomicMax(out + 3, v.w);
}

// h = relu(decode(agg) + b1)
__global__ __launch_bounds__(256) void bias_relu(
        const unsigned* __restrict__ agg, const float* __restrict__ b,
        float* __restrict__ out, int n) {
    int i = blockIdx.x * 256 + threadIdx.x;
    if (i < n) {
        float v = ord2f(agg[i]) + b[i & 127];
        out[i] = v > 0.0f ? v : 0.0f;
    }
}

// out = tanh(x[:, :128] + decode(agg) + b2)
__global__ __launch_bounds__(256) void residual_tanh(
        const float* __restrict__ x, const unsigned* __restrict__ agg,
        const float* __restrict__ b, float* __restrict__ out, int n) {
    int i = blockIdx.x * 256 + threadIdx.x;
    if (i < n) {
        int row = i >> 7, f = i & 127;
        out[i] = tanhf(x[(size_t)row * 130 + f] + ord2f(agg[i]) + b[f]);
    }
}

extern "C" void kernel_launch(void* const* d_in, const int* in_sizes, int n_in,
                              void* d_out, int out_size, void* d_ws, size_t ws_size,
                              hipStream_t stream) {
    const float* x  = (const float*)d_in[0];   // N x 130
    const int*   ei = (const int*)  d_in[1];   // 2 x E (src row, then dst row)
    const float* W1 = (const float*)d_in[2];   // 130 x 128
    const float* b1 = (const float*)d_in[3];   // 128
    const float* W2 = (const float*)d_in[4];   // 128 x 128
    const float* b2 = (const float*)d_in[5];   // 128

    const int K1 = 130, F = 128;
    const int N = in_sizes[0] / K1;
    const int E = in_sizes[1] / 2;
    const int n_feat = N * F;

    float*    buf0 = (float*)d_ws;             // N x 128
    float*    buf1 = buf0 + (size_t)n_feat;    // N x 128
    unsigned* u0   = (unsigned*)buf0;
    unsigned* u1   = (unsigned*)buf1;

    dim3 blk(256);
    dim3 gemmGrid((((N + 15) / 16) + 7) / 8);  // 8 waves (16-row slabs) / block
    dim3 ewGrid((n_feat + 255) / 256);
    dim3 edgeGrid((E + 7) / 8);                // 8 edges (waves) / block

    // ---- conv1: h = relu(segmax((x@W1)[src] -> dst, +self) + b1) ----
    gemm_wmma_f32   <<<gemmGrid, blk, 0, stream>>>(x, W1, buf0, N, K1);
    encode_self     <<<ewGrid,   blk, 0, stream>>>(buf0, u0, u1, n_feat);
    scatter_max_kernel<<<edgeGrid, blk, 0, stream>>>(u0, ei, ei + E, u1, E);
    bias_relu       <<<ewGrid,   blk, 0, stream>>>(u1, b1, buf0, n_feat);

    // ---- conv_out: h2 = segmax((h@W2)[src] -> dst, +self) + b2 ----
    gemm_wmma_f32   <<<gemmGrid, blk, 0, stream>>>(buf0, W2, buf1, N, F);
    encode_self     <<<ewGrid,   blk, 0, stream>>>(buf1, u1, u0, n_feat);
    scatter_max_kernel<<<edgeGrid, blk, 0, stream>>>(u1, ei, ei + E, u0, E);

    // ---- out = tanh(x[:, :128] + h2) ----
    residual_tanh   <<<ewGrid,   blk, 0, stream>>>(x, u0, b2, (float*)d_out, n_feat);
}